// ParametricLIF_523986010138
// MI455X (gfx1250) — compile-verified
//
#include <hip/hip_runtime.h>

// Parametric LIF forward scan for MI455X (gfx1250, wave32).
// x: [N=32, T=64, D=32768] f32; out: spikes, same shape, f32.
// Pure HBM-streaming problem: b128 non-temporal loads/stores, software-
// pipelined t-loop (batches of 8 float4 with register double-buffering).

typedef float v4f __attribute__((ext_vector_type(4)));

constexpr int PLIF_N  = 32;
constexpr int PLIF_T  = 64;
constexpr int PLIF_D  = 32768;
constexpr int PLIF_DV = PLIF_D / 4;   // 8192 float4 per (n, t) row
constexpr int BATCH   = 8;            // float4 timesteps in flight per wave

__global__ __launch_bounds__(256)
void plif_fwd_kernel(const float* __restrict__ x,
                     const float* __restrict__ tau_w,
                     const float* __restrict__ th,
                     float* __restrict__ out) {
    const int tid = blockIdx.x * blockDim.x + threadIdx.x;   // 0 .. N*DV-1
    const int n   = tid >> 13;            // tid / DV  (DV = 8192 = 2^13)
    const int dv  = tid & (PLIF_DV - 1);  // tid % DV

    // k = sigmoid(tau_w); scalar per thread (uniform), negligible cost.
    const float k   = 1.0f / (1.0f + __expf(-tau_w[0]));
    const float thv = th[0];

    const size_t base = (size_t)n * (size_t)(PLIF_T * PLIF_DV) + (size_t)dv;
    const v4f* __restrict__ xp = reinterpret_cast<const v4f*>(x) + base;
    v4f* __restrict__ op       = reinterpret_cast<v4f*>(out) + base;

    v4f v = {0.0f, 0.0f, 0.0f, 0.0f};   // membrane potential (4 lanes of d)

    // Prime the pipeline: batch 0 loads (non-temporal, th:NT).
    v4f cur[BATCH];
#pragma unroll
    for (int i = 0; i < BATCH; ++i)
        cur[i] = __builtin_nontemporal_load(xp + (size_t)i * PLIF_DV);

#pragma unroll
    for (int t0 = 0; t0 < PLIF_T; t0 += BATCH) {
        const bool more = (t0 + BATCH) < PLIF_T;   // compile-time per iteration
        v4f nxt[BATCH];
        if (more) {
            // Issue next batch's loads before touching the serial chain:
            // keeps ~8 b128 loads outstanding per wave over the 128 KB-strided
            // t-walk, hiding HBM latency on top of wave-level parallelism.
#pragma unroll
            for (int i = 0; i < BATCH; ++i)
                nxt[i] = __builtin_nontemporal_load(
                    xp + (size_t)(t0 + BATCH + i) * PLIF_DV);
        }

#pragma unroll
        for (int i = 0; i < BATCH; ++i) {
            v4f s;
#pragma unroll
            for (int c = 0; c < 4; ++c) {
                // h = v + k*(x - v)
                const float h  = __builtin_fmaf(k, cur[i][c] - v[c], v[c]);
                const bool  sp = (h >= thv);       // Heaviside fire
                s[c] = sp ? 1.0f : 0.0f;
                v[c] = sp ? 0.0f : h;              // hard reset
            }
            __builtin_nontemporal_store(s, op + (size_t)(t0 + i) * PLIF_DV);
        }

        if (more) {
#pragma unroll
            for (int i = 0; i < BATCH; ++i)
                cur[i] = nxt[i];                   // SSA-renamed away
        }
    }
}

extern "C" void kernel_launch(void* const* d_in, const int* in_sizes, int n_in,
                              void* d_out, int out_size, void* d_ws, size_t ws_size,
                              hipStream_t stream) {
    (void)in_sizes; (void)n_in; (void)out_size; (void)d_ws; (void)ws_size;
    const float* x     = (const float*)d_in[0];
    const float* tau_w = (const float*)d_in[1];
    const float* th    = (const float*)d_in[2];
    float*       out   = (float*)d_out;

    const int total_threads = PLIF_N * PLIF_DV;   // 262,144
    const int block = 256;                        // 8 wave32 per block
    const int grid  = (total_threads + block - 1) / block;  // 1024
    plif_fwd_kernel<<<grid, block, 0, stream>>>(x, tau_w, th, out);
}